// testnet_85813446574802
// MI455X (gfx1250) — compile-verified
//
#include <hip/hip_runtime.h>
#include <hip/hip_bf16.h>

// Problem constants (fixed by the reference)
#define B_      4
#define T_      48
#define H_      512
#define W_      512
#define HW_     (H_ * W_)          // 262144
#define CH_     3

// Launch config: each thread owns 4 consecutive pixels (one float4)
#define THREADS 256
#define WAVES_  (THREADS / 32)     // 8 waves per block
#define CHUNK   6                  // event slices per async chunk
#define CHUNKS  (T_ / CHUNK)       // 8 chunks
static_assert(T_ % CHUNK == 0, "chunking must tile T");

// ---- CDNA5 feature detection (device pass only) ----
#if defined(__HIP_DEVICE_COMPILE__) && defined(__gfx1250__) && \
    __has_builtin(__builtin_amdgcn_global_load_async_to_lds_b128) && \
    __has_builtin(__builtin_amdgcn_s_wait_asynccnt)
#define USE_ASYNC 1
#else
#define USE_ASYNC 0
#endif

#if defined(__HIP_DEVICE_COMPILE__) && __has_builtin(__builtin_amdgcn_exp2f)
#define EXP2F(x) __builtin_amdgcn_exp2f(x)
#else
#define EXP2F(x) exp2f(x)
#endif

#if USE_ASYNC
// Builtin signature (from hipcc diagnostic): param0 is
//   'int __vector(4) __device__ *'  == AS1 (global) pointer to int4
// so pass exact vector-pointer types in the right address spaces.
typedef int v4i_ __attribute__((vector_size(16)));

// One lane's 16B: global -> LDS, async (tracked by ASYNCcnt)
__device__ __forceinline__ void async_ld16(const float* g, float4* l) {
    __builtin_amdgcn_global_load_async_to_lds_b128(
        (__attribute__((address_space(1))) v4i_*)(const_cast<float*>(g)),
        (__attribute__((address_space(3))) v4i_*)l,
        /*imm offset*/ 0, /*cpol*/ 0);
}
#endif

__global__ __launch_bounds__(THREADS) void fused_event_exp_norm(
    const float* __restrict__ event,   // [B,T,H,W]
    const float* __restrict__ img,     // [B,3,H,W]
    const float* __restrict__ cp,      // scalar
    const int*  __restrict__ idxp,     // scalar
    float* __restrict__ out)           // [B,3,H,W]
{
    const int  tid  = blockIdx.x * THREADS + threadIdx.x;
    const long p    = (long)tid * 4;            // first pixel of this thread's quad
    const int  b    = (int)(p / HW_);
    const int  r    = (int)(p % HW_);           // multiple of 4 -> 16B aligned
    const float k   = (*cp) * 1.44269504088896340736f;  // c * log2(e)
    const int  idxv = *idxp;

    const float* evb = event + (size_t)b * (size_t)(T_ * HW_) + r;

    // running prefix sums, acc = sum_j exp2(k*S[j]) (j=0 term = 1), base = exp2(k*S[idx])
    float sx = 0.f, sy = 0.f, sz = 0.f, sw = 0.f;
    float ax = 1.f, ay = 1.f, az = 1.f, aw = 1.f;
    float bx = 1.f, by = 1.f, bz = 1.f, bw = 1.f;

#if USE_ASYNC
    // Per-wave disjoint double buffers: no block barriers needed.
    __shared__ float4 lds[WAVES_][2][CHUNK][32];   // 48 KB/block
    const int wave = threadIdx.x >> 5;
    const int lane = threadIdx.x & 31;

    // Prime the pipeline: chunks 0 and 1 in flight (12 async loads / wave)
    #pragma unroll
    for (int sl = 0; sl < CHUNK; ++sl)
        async_ld16(evb + (size_t)(0 * CHUNK + sl) * HW_, &lds[wave][0][sl][lane]);
    #pragma unroll
    for (int sl = 0; sl < CHUNK; ++sl)
        async_ld16(evb + (size_t)(1 * CHUNK + sl) * HW_, &lds[wave][1][sl][lane]);

    #pragma unroll
    for (int ck = 0; ck < CHUNKS; ++ck) {
        // Async loads complete in order: <=CHUNK outstanding => chunk ck landed.
        if (ck < CHUNKS - 1) __builtin_amdgcn_s_wait_asynccnt(CHUNK);
        else                 __builtin_amdgcn_s_wait_asynccnt(0);

        #pragma unroll
        for (int sl = 0; sl < CHUNK; ++sl) {
            const float4 ev = lds[wave][ck & 1][sl][lane];   // ds_load_b128
            sx += ev.x; sy += ev.y; sz += ev.z; sw += ev.w;
            const float ex = EXP2F(k * sx), ey = EXP2F(k * sy),
                        ez = EXP2F(k * sz), ew = EXP2F(k * sw);
            ax += ex; ay += ey; az += ez; aw += ew;
            const int j = ck * CHUNK + sl + 1;               // S[j]
            if (j == idxv) { bx = ex; by = ey; bz = ez; bw = ew; }
        }

        // Refill the buffer we just drained (compiler's dscnt waits on the
        // consumed ds_loads guarantee the WAR ordering vs. these async writes).
        if (ck + 2 < CHUNKS) {
            const int nk = ck + 2;
            #pragma unroll
            for (int sl = 0; sl < CHUNK; ++sl)
                async_ld16(evb + (size_t)(nk * CHUNK + sl) * HW_,
                           &lds[wave][nk & 1][sl][lane]);
        }
    }
#else
    // Fallback: direct coalesced float4 global loads
    #pragma unroll
    for (int t = 0; t < T_; ++t) {
        const float4 ev = *reinterpret_cast<const float4*>(evb + (size_t)t * HW_);
        sx += ev.x; sy += ev.y; sz += ev.z; sw += ev.w;
        const float ex = EXP2F(k * sx), ey = EXP2F(k * sy),
                    ez = EXP2F(k * sz), ew = EXP2F(k * sw);
        ax += ex; ay += ey; az += ez; aw += ew;
        if (t + 1 == idxv) { bx = ex; by = ey; bz = ez; bw = ew; }
    }
#endif

    // out = img / E ; E = acc * exp2(-k*S[idx]) / (T+1)  =>  out = img * (T+1)*base/acc
    const float fx = (float)(T_ + 1) * bx / ax;
    const float fy = (float)(T_ + 1) * by / ay;
    const float fz = (float)(T_ + 1) * bz / az;
    const float fw = (float)(T_ + 1) * bw / aw;

    const size_t ioff = (size_t)b * (size_t)(CH_ * HW_) + r;
    #pragma unroll
    for (int ch = 0; ch < CH_; ++ch) {
        const float4 iv = *reinterpret_cast<const float4*>(img + ioff + (size_t)ch * HW_);
        float4 ov;
        ov.x = iv.x * fx; ov.y = iv.y * fy; ov.z = iv.z * fz; ov.w = iv.w * fw;
        *reinterpret_cast<float4*>(out + ioff + (size_t)ch * HW_) = ov;
    }
}

extern "C" void kernel_launch(void* const* d_in, const int* in_sizes, int n_in,
                              void* d_out, int out_size, void* d_ws, size_t ws_size,
                              hipStream_t stream) {
    const float* event = (const float*)d_in[0];   // [4,48,512,512] f32
    const float* img   = (const float*)d_in[1];   // [4,3,512,512]  f32
    const float* c     = (const float*)d_in[2];   // scalar f32
    const int*   idx   = (const int*)d_in[3];     // scalar int
    float* out = (float*)d_out;                   // [4,3,512,512]  f32

    const int total_quads = (B_ * HW_) / 4;       // 262144 threads
    dim3 grid(total_quads / THREADS), block(THREADS);
    fused_event_exp_norm<<<grid, block, 0, stream>>>(event, img, c, idx, out);

    (void)in_sizes; (void)n_in; (void)out_size; (void)d_ws; (void)ws_size;
}